// GIN_74483322847411
// MI455X (gfx1250) — compile-verified
//
#include <hip/hip_runtime.h>

typedef __attribute__((ext_vector_type(2))) float v2f;
typedef __attribute__((ext_vector_type(8))) float v8f;

#define GIN_N_NODES 50000
#define GIN_N_EDGES 640000

// ---------------------------------------------------------------------------
// Scatter-add aggregation: agg[dst[e]][:] += x[src[e]][:]
// One thread per (edge, 4-feature chunk). For D=128 one wave covers one edge
// (32 lanes x float4): src/dst index loads are wave-uniform, float4 gather
// fully coalesced. This phase is the HBM-roofline bottleneck (~2 GB total).
// ---------------------------------------------------------------------------
__global__ void __launch_bounds__(256) gin_scatter_add(
    const float* __restrict__ x, const int* __restrict__ src,
    const int* __restrict__ dst, float* __restrict__ agg, int d4, int D)
{
    long tid = (long)blockIdx.x * blockDim.x + threadIdx.x;
    long total = (long)GIN_N_EDGES * d4;
    if (tid >= total) return;
    int e = (int)(tid / d4);
    int c = (int)(tid % d4) * 4;
    int s = src[e];
    int d = dst[e];
    const float4 v = *reinterpret_cast<const float4*>(x + (long)s * D + c);
    float* p = agg + (long)d * D + c;
    atomicAdd(p + 0, v.x);
    atomicAdd(p + 1, v.y);
    atomicAdd(p + 2, v.z);
    atomicAdd(p + 3, v.w);
}

// ---------------------------------------------------------------------------
// Fused (A0 [+ A1]) @ W + bias [-> ReLU] GEMM using V_WMMA_F32_16X16X4_F32.
//
// Block = 256 threads = 8 waves; block computes MT=4 M-tiles (64 rows) across
// all N columns. A (64 x K, with the GIN "x + agg" add fused) is staged in
// LDS with +4-float row padding (row stride % 64 = 4 -> the 16-rows-same-
// column fragment reads are bank-conflict-free). Each wave owns N/16/8
// column tiles; every B fragment (2 global b32 loads) is reused across the
// 4 M-tile accumulators -> ~4 WMMAs per 2 B loads instead of 1:2.
//
// f32 WMMA fragment layout (ISA 7.12.2):
//   A 16x4 : lane L<16 -> M=L, regs {K=0,K=1}; lane L>=16 -> M=L-16, {K=2,K=3}
//   B 4x16 : mirrored (lanes = N, regs = K pair selected by lane/16)
//   C/D    : reg r -> row r (lanes 0-15) / row r+8 (lanes 16-31), col = lane%16
// ---------------------------------------------------------------------------
template<int K, int N, bool RELU, bool ADD>
__global__ void __launch_bounds__(256) gin_gemm_wmma(
    const float* __restrict__ A0, const float* __restrict__ A1,
    const float* __restrict__ W, const float* __restrict__ bias,
    float* __restrict__ out, int M)
{
    constexpr int MT  = 4;                   // M-tiles (of 16 rows) per block
    constexpr int LDA = K + 4;               // LDS row padding
    __shared__ float lds_a[MT * 16 * LDA];   // K=256: 66.5 KB (<320 KB/WGP)

    const int tid = threadIdx.x;
    const int m0  = blockIdx.x * (MT * 16);

    // Cooperative stage of the 64 x K A tile, fusing the GIN add.
    for (int i = tid; i < MT * 16 * K; i += 256) {
        int r = i / K;
        int c = i - r * K;
        float v = 0.0f;
        if (m0 + r < M) {
            long gi = (long)(m0 + r) * K + c;
            v = A0[gi];
            if (ADD) v += A1[gi];
        }
        lds_a[r * LDA + c] = v;
    }
    __syncthreads();

    const int wave = tid >> 5;
    const int lane = tid & 31;
    const int half = lane >> 4;              // 0 or 1 (selects K pair)
    const int l16  = lane & 15;
    constexpr int TPW = (N / 16) / 8;        // N-tiles per wave (8 waves/block)
    static_assert(TPW >= 1, "need >= 8 column tiles");

    v8f acc[TPW][MT];
    int n0[TPW];
    #pragma unroll
    for (int t = 0; t < TPW; ++t) {
        n0[t] = (wave * TPW + t) * 16;
        float bv = bias[n0[t] + l16];        // bias depends on column only
        #pragma unroll
        for (int mt = 0; mt < MT; ++mt)
            #pragma unroll
            for (int r = 0; r < 8; ++r) acc[t][mt][r] = bv;
    }

    const int acol = half * 2;               // K pair within a 4-wide step
    for (int k0 = 0; k0 < K; k0 += 4) {
        v2f a[MT];
        #pragma unroll
        for (int mt = 0; mt < MT; ++mt) {
            const float* ap = lds_a + (mt * 16 + l16) * LDA + k0 + acol;
            a[mt].x = ap[0];                 // A[M][k0 + 2*half + 0]
            a[mt].y = ap[1];                 // A[M][k0 + 2*half + 1]
        }
        #pragma unroll
        for (int t = 0; t < TPW; ++t) {
            const float* wp = W + (long)(k0 + acol) * N + n0[t] + l16;
            v2f b;
            b.x = wp[0];                     // B[k0 + 2*half + 0][n]
            b.y = wp[N];                     // B[k0 + 2*half + 1][n]
            #pragma unroll
            for (int mt = 0; mt < MT; ++mt) {
                acc[t][mt] = __builtin_amdgcn_wmma_f32_16x16x4_f32(
                    false, a[mt], false, b, (short)0, acc[t][mt], false, false);
            }
        }
    }

    #pragma unroll
    for (int t = 0; t < TPW; ++t) {
        #pragma unroll
        for (int mt = 0; mt < MT; ++mt) {
            const int gr0 = m0 + mt * 16 + half * 8;
            #pragma unroll
            for (int r = 0; r < 8; ++r) {
                float v = acc[t][mt][r];
                if (RELU) v = fmaxf(v, 0.0f);
                int gr = gr0 + r;
                if (gr < M) out[(long)gr * N + n0[t] + l16] = v;
            }
        }
    }
}

// ---------------------------------------------------------------------------
// Launch: layer1 (scatter D=128, gemm 128->256 relu, gemm 256->256 + outer
// relu), layer2 (scatter D=256, gemm 256->256 relu, gemm 256->128).
// ---------------------------------------------------------------------------
extern "C" void kernel_launch(void* const* d_in, const int* in_sizes, int n_in,
                              void* d_out, int out_size, void* d_ws, size_t ws_size,
                              hipStream_t stream)
{
    (void)in_sizes; (void)n_in; (void)out_size; (void)ws_size;

    const float* x   = (const float*)d_in[0];
    const int*   ei  = (const int*)d_in[1];
    const int*   src = ei;
    const int*   dst = ei + GIN_N_EDGES;
    const float* W1a = (const float*)d_in[2];
    const float* b1a = (const float*)d_in[3];
    const float* W1b = (const float*)d_in[4];
    const float* b1b = (const float*)d_in[5];
    const float* W2a = (const float*)d_in[6];
    const float* b2a = (const float*)d_in[7];
    const float* W2b = (const float*)d_in[8];
    const float* b2b = (const float*)d_in[9];
    float* out = (float*)d_out;

    float* ws  = (float*)d_ws;
    float* agg = ws;                                   // 50000*256 f32
    float* h1  = ws + (size_t)GIN_N_NODES * 256;       // 50000*256 f32
    float* h   = h1 + (size_t)GIN_N_NODES * 256;       // 50000*256 f32
    float* g   = h1;                                   // h1 dead after gemm1b

    const int MB = (GIN_N_NODES + 63) / 64;            // 782 blocks (64 rows each)

    // ---- layer 1 ----
    hipMemsetAsync(agg, 0, (size_t)GIN_N_NODES * 128 * sizeof(float), stream);
    {
        long total = (long)GIN_N_EDGES * 32;           // D=128 -> 32 chunks
        int blocks = (int)((total + 255) / 256);
        gin_scatter_add<<<blocks, 256, 0, stream>>>(x, src, dst, agg, 32, 128);
    }
    gin_gemm_wmma<128, 256, true,  true ><<<MB, 256, 0, stream>>>(x,  agg, W1a, b1a, h1, GIN_N_NODES);
    gin_gemm_wmma<256, 256, true,  false><<<MB, 256, 0, stream>>>(h1, h1,  W1b, b1b, h,  GIN_N_NODES);

    // ---- layer 2 ----
    hipMemsetAsync(agg, 0, (size_t)GIN_N_NODES * 256 * sizeof(float), stream);
    {
        long total = (long)GIN_N_EDGES * 64;           // D=256 -> 64 chunks
        int blocks = (int)((total + 255) / 256);
        gin_scatter_add<<<blocks, 256, 0, stream>>>(h, src, dst, agg, 64, 256);
    }
    gin_gemm_wmma<256, 256, true,  true ><<<MB, 256, 0, stream>>>(h, agg, W2a, b2a, g,   GIN_N_NODES);
    gin_gemm_wmma<256, 128, false, false><<<MB, 256, 0, stream>>>(g, g,   W2b, b2b, out, GIN_N_NODES);
}